// DifferentiablePortfolioSim_40802189312499
// MI455X (gfx1250) — compile-verified
//
#include <hip/hip_runtime.h>
#include <hip/hip_bf16.h>

// MI455X / gfx1250, wave32. Memory-bound problem (~1.07GB reads @ 23.3TB/s => ~47us floor).
// Kernel 1: async global->LDS staging + V_WMMA_F32_16X16X4_F32 (B=ones) row-sum reductions.
// Kernel 2: wave32 shfl-based multiplicative scan (cumprod) per batch row.

typedef __attribute__((ext_vector_type(2))) float v2f;
typedef __attribute__((ext_vector_type(8))) float v8f;

#define TCOST   0.001f
#define B_      1024
#define T_      2048
#define A_      64
#define CH      16          // timesteps per wave
#define NCHUNK  (T_ / CH)   // 128
#define STRIDE  68          // padded LDS row stride (floats): bank = (4*row + col) % 64 -> conflict-free

__device__ __forceinline__ unsigned lds_off(const void* p) {
  // LDS flat addresses carry the aperture tag in [63:32]; [31:0] is the LDS byte offset.
  return (unsigned)(unsigned long long)p;
}

__global__ __launch_bounds__(32)
void portfolio_net_kernel(const float* __restrict__ alloc,
                          const float* __restrict__ rets,
                          float* __restrict__ out_net) {
  __shared__ float smA[18 * STRIDE]; // slot 0 = prev row (t0-1 or zeros), slots 1..16 = t0..t0+15
  __shared__ float smR[16 * STRIDE]; // returns rows t0..t0+15

  const int lane = threadIdx.x;
  const int bid  = blockIdx.x;
  const int b    = bid >> 7;             // bid / NCHUNK
  const int t0   = (bid & (NCHUNK - 1)) * CH;
  const int half = lane >> 4;            // 0 | 1  (K-half / M-row selector)
  const int l16  = lane & 15;            // M row index

  // ---------------- stage returns tile: 16 rows, 8 async B128 ops (2 rows per op) -------------
  {
    const float* gbase = rets + ((size_t)b * T_ + t0) * A_;
#pragma unroll
    for (int i = 0; i < 8; ++i) {
      const int row = 2 * i + half;
      unsigned voff  = (unsigned)(row * (A_ * 4) + l16 * 16);
      unsigned laddr = lds_off(&smR[row * STRIDE + l16 * 4]);
      asm volatile("global_load_async_to_lds_b128 %0, %1, %2"
                   :: "v"(laddr), "v"(voff), "s"(gbase) : "memory");
    }
  }
  // ---------------- stage alloc tile: rows t0-1 .. t0+15 (17 rows), or zeros + 16 rows --------
  if (t0 > 0) {
    const float* gbase = alloc + ((size_t)b * T_ + (t0 - 1)) * A_;
#pragma unroll
    for (int i = 0; i < 9; ++i) {
      const int row = 2 * i + half;      // 0..17
      if (row < 17) {
        unsigned voff  = (unsigned)(row * (A_ * 4) + l16 * 16);
        unsigned laddr = lds_off(&smA[row * STRIDE + l16 * 4]);
        asm volatile("global_load_async_to_lds_b128 %0, %1, %2"
                     :: "v"(laddr), "v"(voff), "s"(gbase) : "memory");
      }
    }
  } else {
    // prev_alloc at t=0 is zeros
    *(v2f*)&smA[lane * 2] = (v2f){0.f, 0.f};
    const float* gbase = alloc + ((size_t)b * T_) * A_;
#pragma unroll
    for (int i = 0; i < 8; ++i) {
      const int row = 2 * i + half;
      unsigned voff  = (unsigned)(row * (A_ * 4) + l16 * 16);
      unsigned laddr = lds_off(&smA[(row + 1) * STRIDE + l16 * 4]);
      asm volatile("global_load_async_to_lds_b128 %0, %1, %2"
                   :: "v"(laddr), "v"(voff), "s"(gbase) : "memory");
    }
  }
  asm volatile("s_wait_asynccnt 0x0" ::: "memory");   // LDS tiles visible to this wave

  // ---------------- A=64 reductions via 16 chained K=4 f32 WMMAs, B = ones --------------------
  // A-matrix layout (16x4 f32): lane L -> M=L%16; VGPR0/1 hold K=2*(L/16), 2*(L/16)+1.
  // D[m][n] = sum_k A[m][k] * 1  => every column holds the K-rowsum (exact fp32 accumulation).
  v8f c_to = {};   // turnover sums  sum_a |a_t - a_{t-1}|
  v8f c_pr = {};   // port-return sums sum_a a_t * r_t
  v2f ones; ones[0] = 1.0f; ones[1] = 1.0f;
  const int m = l16;
#pragma unroll
  for (int k = 0; k < 16; ++k) {
    const int c = 4 * k + 2 * half;                       // asset column pair for this lane
    v2f a = *(const v2f*)&smA[(m + 1) * STRIDE + c];      // alloc[t0+m]
    v2f p = *(const v2f*)&smA[ m      * STRIDE + c];      // alloc[t0+m-1]
    v2f r = *(const v2f*)&smR[ m      * STRIDE + c];      // returns[t0+m]
    v2f d, q;
    d[0] = __builtin_fabsf(a[0] - p[0]);
    d[1] = __builtin_fabsf(a[1] - p[1]);
    q[0] = a[0] * r[0];
    q[1] = a[1] * r[1];
    c_to = __builtin_amdgcn_wmma_f32_16x16x4_f32(false, d, false, ones, (short)0, c_to, false, false);
    c_pr = __builtin_amdgcn_wmma_f32_16x16x4_f32(false, q, false, ones, (short)0, c_pr, false, false);
  }

  // D layout: VGPR r holds M=r (lanes 0-15) and M=8+r (lanes 16-31); columns are redundant.
  v8f nets;
#pragma unroll
  for (int r = 0; r < 8; ++r) nets[r] = c_pr[r] - TCOST * c_to[r];

  if (l16 == 0) {           // lane 0 -> rows 0..7, lane 16 -> rows 8..15
    float* pn = out_net + (size_t)b * T_ + t0 + half * 8;
    *(float4*)(pn)     = make_float4(nets[0], nets[1], nets[2], nets[3]);
    *(float4*)(pn + 4) = make_float4(nets[4], nets[5], nets[6], nets[7]);
  }
}

// ---------------- cumprod: one wave per batch row, 32-wide multiplicative scan ---------------
__global__ __launch_bounds__(32)
void portfolio_equity_kernel(const float* __restrict__ net, float* __restrict__ eq) {
  const int b    = blockIdx.x;
  const int lane = threadIdx.x;
  const float* nb = net + (size_t)b * T_;
  float*       eb = eq  + (size_t)b * (T_ + 1);
  if (lane == 0) eb[0] = 1.0f;
  float carry = 1.0f;
  for (int t0 = 0; t0 < T_; t0 += 32) {
    float x = 1.0f + nb[t0 + lane];
#pragma unroll
    for (int off = 1; off < 32; off <<= 1) {
      float y = __shfl_up(x, off, 32);
      if (lane >= off) x *= y;
    }
    eb[t0 + lane + 1] = carry * x;
    carry *= __shfl(x, 31, 32);
  }
}

extern "C" void kernel_launch(void* const* d_in, const int* in_sizes, int n_in,
                              void* d_out, int out_size, void* d_ws, size_t ws_size,
                              hipStream_t stream) {
  const float* alloc = (const float*)d_in[0];
  const float* rets  = (const float*)d_in[1];
  float* out     = (float*)d_out;
  float* out_eq  = out;                              // [B, T+1]
  float* out_net = out + (size_t)B_ * (T_ + 1);      // [B, T]

  portfolio_net_kernel<<<B_ * NCHUNK, 32, 0, stream>>>(alloc, rets, out_net);
  portfolio_equity_kernel<<<B_, 32, 0, stream>>>(out_net, out_eq);
}